// custom_gnn_3831110828325
// MI455X (gfx1250) — compile-verified
//
#include <hip/hip_runtime.h>
#include <hip/hip_bf16.h>

#define N_NODES 50000
#define N_EDGES 200000
#define DIM     384

typedef __attribute__((ext_vector_type(2))) float v2f;
typedef __attribute__((ext_vector_type(8))) float v8f;

// ---- ordered-uint encoding of float for atomicMax-based segment max ----
__device__ __forceinline__ unsigned f2ord(float f) {
  unsigned u = __float_as_uint(f);
  return (u & 0x80000000u) ? ~u : (u | 0x80000000u);
}
__device__ __forceinline__ float ord2f(unsigned u) {
  return (u & 0x80000000u) ? __uint_as_float(u & 0x7FFFFFFFu)
                           : __uint_as_float(~u);
}

// ---------------------------------------------------------------------------
// Kernel 0: zero the segment-max (encoded -inf == 0) and segment-sum buffers.
// ---------------------------------------------------------------------------
__global__ void init_kernel(unsigned* __restrict__ m_enc, float* __restrict__ denom, int n) {
  int i = blockIdx.x * blockDim.x + threadIdx.x;
  if (i < n) { m_enc[i] = 0u; denom[i] = 0.0f; }
}

// ---------------------------------------------------------------------------
// Kernel 1: fused X @ [Wq | Wk | Wv | Wskip]  via V_WMMA_F32_16X16X4_F32.
// One wave per 16x16 output tile; K-loop of 96 WMMA steps.
// Skip projection is written straight into d_out (it is the accumulator base).
// ---------------------------------------------------------------------------
__global__ __launch_bounds__(32) void fused_qkvs_gemm(
    const float* __restrict__ X,
    const float* __restrict__ Wq, const float* __restrict__ bq,
    const float* __restrict__ Wk, const float* __restrict__ bk,
    const float* __restrict__ Wv, const float* __restrict__ bv,
    const float* __restrict__ Ws, const float* __restrict__ bs,
    float* __restrict__ qo, float* __restrict__ ko,
    float* __restrict__ vo, float* __restrict__ skipo)
{
  const int lane = threadIdx.x;
  const int half = lane >> 4;      // 0: K pair {0,1}, 1: K pair {2,3}
  const int lid  = lane & 15;      // row (A) / col (B,C,D) within tile
  const int r0   = blockIdx.x * 16;
  const int ct   = blockIdx.y;     // 0..95 column tiles of the 1536-wide fused W
  const int wsel = ct / 24;        // 24 16-col tiles per 384-col weight matrix
  const int cb   = (ct % 24) * 16; // column base inside the selected weight

  const float* W; const float* B; float* O;
  switch (wsel) {
    case 0:  W = Wq; B = bq; O = qo;    break;
    case 1:  W = Wk; B = bk; O = ko;    break;
    case 2:  W = Wv; B = bv; O = vo;    break;
    default: W = Ws; B = bs; O = skipo; break;
  }

  v8f acc = {};
  const float* xrow = X + (size_t)(r0 + lid) * DIM + 2 * half;        // A: row r0+lid
  const float* wcol = W + (size_t)(2 * half) * DIM + cb + lid;        // B: col cb+lid

  #pragma unroll 4
  for (int k0 = 0; k0 < DIM; k0 += 4) {
    float2 af = *(const float2*)(xrow + k0);   // X[row, k0+2h .. k0+2h+1]
    v2f a, b;
    a.x = af.x;
    a.y = af.y;
    b.x = wcol[(size_t)k0 * DIM];              // W[k0+2h,   col]
    b.y = wcol[(size_t)k0 * DIM + DIM];        // W[k0+2h+1, col]
    acc = __builtin_amdgcn_wmma_f32_16x16x4_f32(
        /*neg_a=*/false, a, /*neg_b=*/false, b,
        /*c_mod=*/(short)0, acc, /*reuse_a=*/false, /*reuse_b=*/false);
  }

  const float bias = B[cb + lid];
  #pragma unroll
  for (int i = 0; i < 8; ++i) {
    const int m = i + 8 * half;                // C/D layout: M = vgpr + 8*(lane/16)
    O[(size_t)(r0 + m) * DIM + cb + lid] = acc[i] + bias;
  }
}

// ---------------------------------------------------------------------------
// Kernel 2: per-edge attention logit + scatter-max. One wave32 per edge,
// 12 channels per lane, wave-level shuffle reduction.
// ---------------------------------------------------------------------------
__global__ __launch_bounds__(256) void edge_alpha_kernel(
    const long long* __restrict__ src, const long long* __restrict__ dst,
    const float* __restrict__ ea,
    const float* __restrict__ q, const float* __restrict__ k,
    const float* __restrict__ We, const float* __restrict__ be,
    float* __restrict__ alpha, unsigned* __restrict__ m_enc)
{
  const int e    = (blockIdx.x * blockDim.x + threadIdx.x) >> 5;
  const int lane = threadIdx.x & 31;
  if (e >= N_EDGES) return;

  const long long s = src[e], d = dst[e];
  const float eaj = ea[e];
  const float* qrow = q + (size_t)d * DIM;
  const float* krow = k + (size_t)s * DIM;

  float sum = 0.0f;
  #pragma unroll
  for (int t = 0; t < DIM / 32; ++t) {
    const int c = lane + 32 * t;
    const float ef = fmaf(eaj, We[c], be[c]);      // e[j,c] = ea*We[c] + be[c]
    sum = fmaf(qrow[c], krow[c] + ef, sum);
  }
  #pragma unroll
  for (int off = 16; off > 0; off >>= 1) sum += __shfl_xor(sum, off, 32);

  if (lane == 0) {
    const float a = sum * 0.05103103630798288f;    // 1/sqrt(384)
    alpha[e] = a;
    atomicMax(m_enc + d, f2ord(a));                // global_atomic_max_u32
  }
}

// ---------------------------------------------------------------------------
// Kernel 3: ex = exp(alpha - m[dst]); denom[dst] += ex  (hw fp32 atomic)
// ---------------------------------------------------------------------------
__global__ __launch_bounds__(256) void edge_exp_kernel(
    const long long* __restrict__ dst,
    const float* __restrict__ alpha, const unsigned* __restrict__ m_enc,
    float* __restrict__ ex, float* __restrict__ denom)
{
  const int j = blockIdx.x * blockDim.x + threadIdx.x;
  if (j >= N_EDGES) return;
  const long long d = dst[j];
  const float m = ord2f(m_enc[d]);
  const float v = __expf(alpha[j] - m);
  ex[j] = v;
  unsafeAtomicAdd(denom + d, v);                   // global_atomic_add_f32
}

// ---------------------------------------------------------------------------
// Kernel 4: out[dst] += (ex/denom[dst]) * (v[src] + e). One wave32 per edge.
// ---------------------------------------------------------------------------
__global__ __launch_bounds__(256) void edge_scatter_kernel(
    const long long* __restrict__ src, const long long* __restrict__ dst,
    const float* __restrict__ ea,
    const float* __restrict__ v, const float* __restrict__ We,
    const float* __restrict__ be,
    const float* __restrict__ ex, const float* __restrict__ denom,
    float* __restrict__ out)
{
  const int e    = (blockIdx.x * blockDim.x + threadIdx.x) >> 5;
  const int lane = threadIdx.x & 31;
  if (e >= N_EDGES) return;

  const long long s = src[e], d = dst[e];
  const float coef = ex[e] / denom[d];
  const float eaj  = ea[e];
  const float* vrow = v + (size_t)s * DIM;
  float* orow = out + (size_t)d * DIM;

  #pragma unroll
  for (int t = 0; t < DIM / 32; ++t) {
    const int c = lane + 32 * t;
    const float ef = fmaf(eaj, We[c], be[c]);
    unsafeAtomicAdd(orow + c, coef * (vrow[c] + ef));  // global_atomic_add_f32
  }
}

// ---------------------------------------------------------------------------
extern "C" void kernel_launch(void* const* d_in, const int* in_sizes, int n_in,
                              void* d_out, int out_size, void* d_ws, size_t ws_size,
                              hipStream_t stream) {
  const float*     X    = (const float*)d_in[0];
  const long long* eidx = (const long long*)d_in[1];   // int64 [2, E]
  const float*     ea   = (const float*)d_in[2];       // [E, 1]
  const float* Wq = (const float*)d_in[3];  const float* bq = (const float*)d_in[4];
  const float* Wk = (const float*)d_in[5];  const float* bk = (const float*)d_in[6];
  const float* Wv = (const float*)d_in[7];  const float* bv = (const float*)d_in[8];
  const float* We = (const float*)d_in[9];  const float* be = (const float*)d_in[10];
  const float* Ws = (const float*)d_in[11]; const float* bs = (const float*)d_in[12];
  float* out = (float*)d_out;

  const long long* src = eidx;             // edge_index[0]
  const long long* dst = eidx + N_EDGES;   // edge_index[1]

  // workspace carve-up (~232 MB)
  char* ws = (char*)d_ws;
  const size_t ncf = (size_t)N_NODES * DIM * sizeof(float);
  float*    q     = (float*)ws;  ws += ncf;
  float*    k     = (float*)ws;  ws += ncf;
  float*    v     = (float*)ws;  ws += ncf;
  float*    alpha = (float*)ws;  ws += (size_t)N_EDGES * sizeof(float);
  float*    ex    = (float*)ws;  ws += (size_t)N_EDGES * sizeof(float);
  unsigned* m_enc = (unsigned*)ws; ws += (size_t)N_NODES * sizeof(unsigned);
  float*    denom = (float*)ws;

  init_kernel<<<(N_NODES + 255) / 256, 256, 0, stream>>>(m_enc, denom, N_NODES);

  dim3 gg(N_NODES / 16, (4 * DIM) / 16);   // 3125 x 96 tiles, 1 wave each
  fused_qkvs_gemm<<<gg, 32, 0, stream>>>(X, Wq, bq, Wk, bk, Wv, bv, Ws, bs,
                                         q, k, v, out);

  const int edgeWaveBlocks = N_EDGES / 8;  // 8 waves (edges) per 256-thread block
  edge_alpha_kernel<<<edgeWaveBlocks, 256, 0, stream>>>(src, dst, ea, q, k, We, be,
                                                        alpha, m_enc);
  edge_exp_kernel<<<(N_EDGES + 255) / 256, 256, 0, stream>>>(dst, alpha, m_enc,
                                                             ex, denom);
  edge_scatter_kernel<<<edgeWaveBlocks, 256, 0, stream>>>(src, dst, ea, v, We, be,
                                                          ex, denom, out);
}